// PWVD_9826885173699
// MI455X (gfx1250) — compile-verified
//
#include <hip/hip_runtime.h>

typedef float v2f __attribute__((ext_vector_type(2)));
typedef float v8f __attribute__((ext_vector_type(8)));

// D(16x16,f32) += A(16x4,f32) * B(4x16,f32)
#define WMMA4(A, B, Cacc) \
  __builtin_amdgcn_wmma_f32_16x16x4_f32(false, (A), false, (B), (short)0, (Cacc), false, false)

constexpr int Hh   = 64;
constexpr int Wd   = 64;
constexpr int Nw   = 7;           // window / DFT size
constexpr int TILE = 16;          // pixels per wave
constexpr int ROWS = TILE * Nw;   // 112 matrix rows per tile (r = a*16 + p)
constexpr int STR  = 8;           // LDS row stride in floats (col 7 = zero pad)
constexpr int SLABW = 24;         // padded slab width (22 used)
constexpr int WAVE_F = ROWS * STR * 3 + Nw * SLABW; // V1 + A2r + A2i + slab (floats)

__device__ __forceinline__ void dft_coef(int b, int n, float& cc, float& ss) {
  if (b < Nw && n < Nw) {
    float ang = -2.0f * 3.14159265358979323846f * (float)(b * n) / 7.0f;
    cc = __cosf(ang);
    ss = __sinf(ang);   // = -sin(theta): imag-part DFT coefficient
  } else {
    cc = 0.0f;
    ss = 0.0f;
  }
}

__global__ __launch_bounds__(128) void pwvd_wmma_kernel(const float* __restrict__ x,
                                                        float* __restrict__ out) {
  __shared__ __align__(16) float lds[4 * WAVE_F];

  const int lane  = threadIdx.x & 31;
  const int wv    = threadIdx.x >> 5;
  const int nIdx  = lane & 15;       // N column of C/D tile
  const int khalf = lane >> 4;       // 0: K lanes {0,1}, 1: K lanes {2,3}
  const int kh2   = khalf * 2;

  const int bc = blockIdx.x / Hh;    // 0..5  (B*C flattened)
  const int i  = blockIdx.x % Hh;    // output row
  const int j0 = wv * TILE;          // first output col of this wave's tile

  float* V1  = &lds[wv * WAVE_F];    // [112][8]  rows r = a*16 + p
  float* A2r = V1 + ROWS * STR;      // [112][8]  rows r = l*16 + p
  float* A2i = A2r + ROWS * STR;
  float* xs  = A2i + ROWS * STR;     // [7][24]   input slab

  __builtin_prefetch(&x[(bc * Hh + i) * Wd], 0, 0);

  // ---- per-lane constant B operands (DFT matrices, K-padded to 8, N-padded to 16)
  v2f bC0, bS0, bC1, bS1;
  {
    const int k0 = 0 + kh2, k1 = 1 + kh2;
    float c, s;
    dft_coef(k0, nIdx, c, s);     bC0.x = c; bS0.x = s;
    dft_coef(k1, nIdx, c, s);     bC0.y = c; bS0.y = s;
    dft_coef(k0 + 4, nIdx, c, s); bC1.x = c; bS1.x = s;  // row 7 -> zero pad
    dft_coef(k1 + 4, nIdx, c, s); bC1.y = c; bS1.y = s;
  }
  v2f bS0n; bS0n.x = -bS0.x; bS0n.y = -bS0.y;
  v2f bS1n; bS1n.x = -bS1.x; bS1n.y = -bS1.y;

  // ---- zero this wave's LDS region (provides K=7 zero padding)
  for (int t = lane; t < WAVE_F; t += 32) V1[t] = 0.0f;
  __syncthreads();

  // ---- load input slab: rows i-3..i+3, cols j0-3..j0+18 (zero outside image)
  if (lane < 22) {
    const int jj  = j0 - 3 + lane;
    const bool jok = (jj >= 0) && (jj < Wd);
    #pragma unroll
    for (int a = 0; a < Nw; ++a) {
      const int ii = i - 3 + a;
      float v = 0.0f;
      if (jok && ii >= 0 && ii < Hh) v = x[(bc * Hh + ii) * Wd + jj];
      xs[a * SLABW + lane] = v;
    }
  }
  __syncthreads();

  // ---- build stage-1 A: V1[a*16+p][b] = win[a][b] * win[6-a][6-b]
  for (int t = lane; t < ROWS * STR; t += 32) {
    const int r = t >> 3, b = t & 7;      // row, col (col 7 = pad)
    const int a = r >> 4, p = r & 15;
    const int bb = (b < 7) ? b : 0;       // clamp for safe LDS read
    const float w1 = xs[a * SLABW + p + bb];
    const float w2 = xs[(6 - a) * SLABW + p + 6 - bb];
    V1[t] = (b < 7) ? (w1 * w2) : 0.0f;
  }
  __syncthreads();

  // ---- stage 1: row DFT.  Block m => a == m, M == p.
  //      D element (M=p, N=l) scatters to A2[(l*16 + p)][a=m]
  const int sb = nIdx * (16 * STR) + khalf * (8 * STR);  // scatter base (floats)
  #pragma unroll
  for (int m = 0; m < 7; ++m) {
    const int row = 16 * m + nIdx;
    const v2f a0 = *(const v2f*)&V1[row * STR + kh2];       // b = kh2, kh2+1
    const v2f a1 = *(const v2f*)&V1[row * STR + 4 + kh2];   // b = 4+kh2, 5+kh2
    v8f Dr = {}; v8f Di = {};
    Dr = WMMA4(a0, bC0, Dr);
    Dr = WMMA4(a1, bC1, Dr);
    Di = WMMA4(a0, bS0, Di);
    Di = WMMA4(a1, bS1, Di);
    if (nIdx < 7) {
      #pragma unroll
      for (int t = 0; t < 8; ++t) {
        A2r[sb + t * STR + m] = Dr[t];
        A2i[sb + t * STR + m] = Di[t];
      }
    }
  }
  __syncthreads();

  // ---- stage 2: column DFT + magnitude + fftshift + store.
  //      Block m => l == m, M == p, col N == k.
  const int ks = (nIdx < 3) ? (nIdx + 4) : (nIdx - 3);     // (k+4)%7 for k<7
  float* outp = out + ((bc * Hh + i) * Wd + j0) * 49 + ks * 7 + khalf * (8 * 49);
  #pragma unroll
  for (int m = 0; m < 7; ++m) {
    const int row = 16 * m + nIdx;
    const v2f r0 = *(const v2f*)&A2r[row * STR + kh2];
    const v2f r1 = *(const v2f*)&A2r[row * STR + 4 + kh2];
    const v2f i0 = *(const v2f*)&A2i[row * STR + kh2];
    const v2f i1 = *(const v2f*)&A2i[row * STR + 4 + kh2];
    v8f Vr = {}; v8f Vi = {};
    Vr = WMMA4(r0, bC0, Vr);
    Vr = WMMA4(r1, bC1, Vr);
    Vr = WMMA4(i0, bS0n, Vr);
    Vr = WMMA4(i1, bS1n, Vr);
    Vi = WMMA4(i0, bC0, Vi);
    Vi = WMMA4(i1, bC1, Vi);
    Vi = WMMA4(r0, bS0, Vi);
    Vi = WMMA4(r1, bS1, Vi);
    constexpr int LS_TAB[7] = {4, 5, 6, 0, 1, 2, 3};       // (m+4)%7
    const int ls = LS_TAB[m];
    if (nIdx < 7) {
      #pragma unroll
      for (int t = 0; t < 8; ++t) {
        const float mag = __builtin_amdgcn_sqrtf(Vr[t] * Vr[t] + Vi[t] * Vi[t]);
        outp[t * 49 + ls] = mag;   // p = t + 8*khalf folded into outp
      }
    }
  }
}

extern "C" void kernel_launch(void* const* d_in, const int* in_sizes, int n_in,
                              void* d_out, int out_size, void* d_ws, size_t ws_size,
                              hipStream_t stream) {
  (void)in_sizes; (void)n_in; (void)d_ws; (void)ws_size; (void)out_size;
  const float* x = (const float*)d_in[0];   // (2,3,64,64) f32
  // d_in[1] is N == 7, baked in as constexpr
  float* out = (float*)d_out;               // (2,3,64,64,7,7) f32
  dim3 grid(2 * 3 * Hh);   // one block per (bc, i) row; 4 waves cover W
  dim3 block(128);
  pwvd_wmma_kernel<<<grid, block, 0, stream>>>(x, out);
}